// SingleVQC_12t6_12077448036886
// MI455X (gfx1250) — compile-verified
//
#include <hip/hip_runtime.h>
#include <hip/hip_bf16.h>

// ---------------------------------------------------------------------------
// 12-qubit VQC simulator for MI455X (gfx1250).
//  - state (4096 complex amps, re/im f32) resident in LDS per block
//  - each Rot layer folded into three 16x16 complex unitaries (kron of 2x2s)
//  - 16x16 complex tile updates via chained V_WMMA_F32_16X16X4_F32
//  - CNOT ring = in-place LDS pair swaps; <Z> via LDS tree reduction
// ---------------------------------------------------------------------------

typedef __attribute__((ext_vector_type(2))) float v2f;
typedef __attribute__((ext_vector_type(8))) float v8f;

#define NQ     12
#define SDIM   4096           // 2^12 amplitudes
#define DEPTH  6
#define NTHR   256

__device__ __forceinline__ v8f wmma_k4(v2f a, v2f b, v8f c) {
  // D = A(16x4,f32) * B(4x16,f32) + C(16x16,f32)
  return __builtin_amdgcn_wmma_f32_16x16x4_f32(
      /*neg_a=*/false, a, /*neg_b=*/false, b,
      /*c_mod=*/(short)0, c, /*reuse_a=*/false, /*reuse_b=*/false);
}

__device__ __forceinline__ v8f vzero8() {
  v8f z;
#pragma unroll
  for (int i = 0; i < 8; ++i) z[i] = 0.0f;
  return z;
}

// newState = U * T where T[k][n] = s[base + k*ks + n]  (16x16 tile, in-place).
// ur/ui/uni = Re(U), Im(U), -Im(U), 16x16 row-major in LDS.
__device__ __forceinline__ void apply_left_tile(
    float* sre, float* sim, const float* ur, const float* ui, const float* uni,
    int base, int ks, int lane) {
  const int n   = lane & 15;         // A row M / B,C column N for this lane
  const int kh  = (lane >> 4) << 1;  // K sub-offset: 0 or 2
  const int hi8 = (lane >> 4) << 3;  // C row offset: 0 or 8
  v8f cr = vzero8(), ci = vzero8();
#pragma unroll
  for (int kb = 0; kb < 16; kb += 4) {
    const int k0 = kb + kh;
    v2f aur  = { ur [n * 16 + k0],  ur [n * 16 + k0 + 1] };  // A[m][k]=U[m][k]
    v2f aui  = { ui [n * 16 + k0],  ui [n * 16 + k0 + 1] };
    v2f auni = { uni[n * 16 + k0],  uni[n * 16 + k0 + 1] };
    v2f br = { sre[base + k0 * ks + n], sre[base + (k0 + 1) * ks + n] };
    v2f bi = { sim[base + k0 * ks + n], sim[base + (k0 + 1) * ks + n] };
    cr = wmma_k4(aur,  br, cr);   // + Ur*Sr
    cr = wmma_k4(auni, bi, cr);   // - Ui*Si
    ci = wmma_k4(aui,  br, ci);   // + Ui*Sr
    ci = wmma_k4(aur,  bi, ci);   // + Ur*Si
  }
#pragma unroll
  for (int v = 0; v < 8; ++v) {
    const int m = v + hi8;
    sre[base + m * ks + n] = cr[v];
    sim[base + m * ks + n] = ci[v];
  }
}

// newState = S * U^T where S[m][k] = s[base + m*16 + k]  (16x16 tile, in-place)
__device__ __forceinline__ void apply_right_tile(
    float* sre, float* sim, const float* ur, const float* ui, const float* uni,
    int base, int lane) {
  const int n   = lane & 15;
  const int kh  = (lane >> 4) << 1;
  const int hi8 = (lane >> 4) << 3;
  v8f cr = vzero8(), ci = vzero8();
#pragma unroll
  for (int kb = 0; kb < 16; kb += 4) {
    const int k0 = kb + kh;
    v2f asr = { sre[base + n * 16 + k0], sre[base + n * 16 + k0 + 1] };  // A=S
    v2f asi = { sim[base + n * 16 + k0], sim[base + n * 16 + k0 + 1] };
    // B[k][n] = (U^T)[k][n] = U[n][k]
    v2f bur  = { ur [n * 16 + k0],  ur [n * 16 + k0 + 1] };
    v2f bui  = { ui [n * 16 + k0],  ui [n * 16 + k0 + 1] };
    v2f buni = { uni[n * 16 + k0],  uni[n * 16 + k0 + 1] };
    cr = wmma_k4(asr, bur,  cr);
    cr = wmma_k4(asi, buni, cr);
    ci = wmma_k4(asr, bui,  ci);
    ci = wmma_k4(asi, bur,  ci);
  }
#pragma unroll
  for (int v = 0; v < 8; ++v) {
    const int m = v + hi8;
    sre[base + m * 16 + n] = cr[v];
    sim[base + m * 16 + n] = ci[v];
  }
}

__global__ __launch_bounds__(NTHR) void vqc_kernel(
    const float* __restrict__ x, const float* __restrict__ th,
    float* __restrict__ out, int layers) {
  __shared__ float sre[SDIM];          // 16 KB  state real
  __shared__ float sim[SDIM];          // 16 KB  state imag
  __shared__ float s_ur [3][256];      // Re(U) per qubit group
  __shared__ float s_uim[3][256];      // Im(U)
  __shared__ float s_uni[3][256];      // -Im(U)
  __shared__ float g2r[NQ][4], g2i[NQ][4];  // per-qubit Rot 2x2
  __shared__ float ang[NQ];            // current angles (input or fed-back <Z>)
  __shared__ float csn[NQ], snn[NQ];   // cos/sin(angle/2)
  __shared__ float red[NQ][NTHR];      // 12 KB reduction scratch

  const int tid  = threadIdx.x;
  const int lane = tid & 31;
  const int wave = tid >> 5;
  const int b    = blockIdx.x;

  if (tid < NQ) ang[tid] = x[b * NQ + tid];
  __syncthreads();

  for (int L = 0; L < layers; ++L) {
    // ---- AngleEmbedding(RY) on |0..0> == real product state ----------------
    if (tid < NQ) {
      csn[tid] = __cosf(0.5f * ang[tid]);
      snn[tid] = __sinf(0.5f * ang[tid]);
    }
    __syncthreads();
    for (int i = tid; i < SDIM; i += NTHR) {
      float a = 1.0f;
#pragma unroll
      for (int p = 0; p < NQ; ++p)  // flat-index bit p <-> qubit (11-p)
        a *= ((i >> p) & 1) ? snn[11 - p] : csn[11 - p];
      sre[i] = a;
      sim[i] = 0.0f;
    }
    __syncthreads();

    // ---- StronglyEntanglingLayers -----------------------------------------
    for (int l = 0; l < DEPTH; ++l) {
      // per-qubit Rot(phi,theta,omega) = RZ(om) RY(th) RZ(phi)
      if (tid < NQ) {
        const float* t3 = th + ((size_t)((L * DEPTH + l) * NQ + tid)) * 3;
        const float phi = t3[0], the = t3[1], om = t3[2];
        const float ch = __cosf(0.5f * the), sh = __sinf(0.5f * the);
        const float ap = 0.5f * (phi + om), am = 0.5f * (phi - om);
        const float cap = __cosf(ap), sap = __sinf(ap);
        const float cam = __cosf(am), sam = __sinf(am);
        g2r[tid][0] =  ch * cap;  g2i[tid][0] = -ch * sap;  // U00
        g2r[tid][1] = -sh * cam;  g2i[tid][1] = -sh * sam;  // U01
        g2r[tid][2] =  sh * cam;  g2i[tid][2] = -sh * sam;  // U10
        g2r[tid][3] =  ch * cap;  g2i[tid][3] =  ch * sap;  // U11
      }
      __syncthreads();

      // kron of 4 gates -> 16x16 complex unitary per qubit group
#pragma unroll
      for (int g = 0; g < 3; ++g) {
        const int r = tid >> 4, c = tid & 15;
        float vr = 1.0f, vi = 0.0f;
#pragma unroll
        for (int j = 0; j < 4; ++j) {   // qubit 4g+j <-> bit (3-j) of r,c
          const int q  = 4 * g + j;
          const int rb = (r >> (3 - j)) & 1;
          const int cb = (c >> (3 - j)) & 1;
          const float er = g2r[q][rb * 2 + cb];
          const float ei = g2i[q][rb * 2 + cb];
          const float nr = vr * er - vi * ei;
          vi = vr * ei + vi * er;
          vr = nr;
        }
        s_ur [g][tid] = vr;
        s_uim[g][tid] = vi;
        s_uni[g][tid] = -vi;
      }
      __syncthreads();

      // group 0 (qubits 0-3, top 4 bits): view (16 x 256), new = U * T
      for (int t = wave; t < 16; t += 8)
        apply_left_tile(sre, sim, s_ur[0], s_uim[0], s_uni[0], t * 16, 256, lane);
      __syncthreads();
      // group 1 (qubits 4-7, bits 7:4): 16 slices of (16 x 16), new = U * M
      for (int t = wave; t < 16; t += 8)
        apply_left_tile(sre, sim, s_ur[1], s_uim[1], s_uni[1], t * 256, 16, lane);
      __syncthreads();
      // group 2 (qubits 8-11, low 4 bits): view (256 x 16), new = S * U^T
      for (int t = wave; t < 16; t += 8)
        apply_right_tile(sre, sim, s_ur[2], s_uim[2], s_uni[2], t * 256, lane);
      __syncthreads();

      // ---- CNOT ring, range r = l % 11 + 1: in-place pair swaps -----------
      const int rr = l % (NQ - 1) + 1;
      for (int q = 0; q < NQ; ++q) {
        const int cb = 11 - q;
        const int tb = 11 - ((q + rr) % NQ);
        const int lo = cb < tb ? cb : tb;
        const int hi = cb < tb ? tb : cb;
#pragma unroll
        for (int wi = 0; wi < 4; ++wi) {               // 1024 pairs / 256 thr
          const unsigned w = (unsigned)(tid + wi * NTHR);
          const unsigned lowm  = w & ((1u << lo) - 1u);
          const unsigned midm  = (w >> lo) & ((1u << (hi - 1 - lo)) - 1u);
          const unsigned highm = w >> (hi - 1);
          unsigned i = lowm | (midm << (lo + 1)) | (highm << (hi + 1));
          i |= (1u << cb);                    // control = 1, target bit = 0
          const unsigned jj = i | (1u << tb); // partner: target bit = 1
          float t0 = sre[i]; sre[i] = sre[jj]; sre[jj] = t0;
          float t1 = sim[i]; sim[i] = sim[jj]; sim[jj] = t1;
        }
        __syncthreads();
      }
    }

    // ---- measurement: <Z_q> = sum_i |psi_i|^2 * (1 - 2*bit_q(i)) ----------
    float acc[NQ];
#pragma unroll
    for (int q = 0; q < NQ; ++q) acc[q] = 0.0f;
    for (int i = tid; i < SDIM; i += NTHR) {
      const float p = sre[i] * sre[i] + sim[i] * sim[i];
#pragma unroll
      for (int q = 0; q < NQ; ++q)
        acc[q] += ((i >> (11 - q)) & 1) ? -p : p;
    }
#pragma unroll
    for (int q = 0; q < NQ; ++q) red[q][tid] = acc[q];
    __syncthreads();
    for (int s = NTHR / 2; s > 0; s >>= 1) {
      if (tid < s) {
#pragma unroll
        for (int q = 0; q < NQ; ++q) red[q][tid] += red[q][tid + s];
      }
      __syncthreads();
    }
    if (tid < NQ) ang[tid] = red[tid][0];   // feed back as next angles
    __syncthreads();
  }

  // out[b, 0..5] = <Z_{3..8}> * (pi - eps_f32)
  const float MULT = (float)(3.14159265358979323846 - 1.1920928955078125e-07);
  if (tid < 6) out[b * 6 + tid] = ang[tid + 3] * MULT;
}

extern "C" void kernel_launch(void* const* d_in, const int* in_sizes, int n_in,
                              void* d_out, int out_size, void* d_ws, size_t ws_size,
                              hipStream_t stream) {
  const float* x  = (const float*)d_in[0];   // (B, 12) f32
  const float* th = (const float*)d_in[1];   // (LAYERS, DEPTH, 12, 3) f32
  float* out = (float*)d_out;                // (B, 6) f32
  const int B = in_sizes[0] / NQ;
  int layers = in_sizes[1] / (DEPTH * NQ * 3);
  if (layers < 1) layers = 1;
  vqc_kernel<<<B, NTHR, 0, stream>>>(x, th, out, layers);
}